// VRWKV_Multi_HW_SpatialMix_12249246728667
// MI455X (gfx1250) — compile-verified
//
#include <hip/hip_runtime.h>
#include <hip/hip_bf16.h>
#include <math.h>

// ---------------------------------------------------------------------------
// Problem constants (from setup_inputs): B=8, h=w=64 -> T=4096, C=256, D=2C=512
// ---------------------------------------------------------------------------
#define BATCH 8
#define HH 64
#define WW 64
#define TT (HH * WW)       // 4096
#define CC 256
#define DD (2 * CC)        // 512
#define NCHUNK 64          // chunks for parallel WKV scan
#define CLEN (TT / NCHUNK) // 64 steps per chunk

typedef float v2f __attribute__((ext_vector_type(2)));
typedef float v8f __attribute__((ext_vector_type(8)));

// ---------------------------------------------------------------------------
// Kernel 1: fold alpha-weighted {identity, 1x1, 3x3, 5x5} depthwise stack into
// one effective 5x5 depthwise kernel per channel.
// ---------------------------------------------------------------------------
__global__ __launch_bounds__(CC) void build_eff_kernel(
    const float* __restrict__ alpha, const float* __restrict__ cw1,
    const float* __restrict__ cw3, const float* __restrict__ cw5,
    float* __restrict__ eff) {
  int c = threadIdx.x;
  float a0 = alpha[0], a1 = alpha[1], a2 = alpha[2], a3 = alpha[3];
  float e[25];
#pragma unroll
  for (int i = 0; i < 25; ++i) e[i] = a3 * cw5[c * 25 + i];
#pragma unroll
  for (int r = 0; r < 3; ++r)
#pragma unroll
    for (int q = 0; q < 3; ++q) e[(r + 1) * 5 + (q + 1)] += a2 * cw3[c * 9 + r * 3 + q];
  e[12] += a1 * cw1[c] + a0;  // 1x1 conv + identity at center tap
#pragma unroll
  for (int i = 0; i < 25; ++i) eff[c * 25 + i] = e[i];
}

// ---------------------------------------------------------------------------
// Kernel 2: effective 5x5 depthwise conv, NCHW semantics on [B,T,C] layout.
// One block per (b,t) pixel, 256 threads = channels (coalesced on C).
// ---------------------------------------------------------------------------
__global__ __launch_bounds__(CC) void omni_conv_kernel(
    const float* __restrict__ x, const float* __restrict__ eff,
    float* __restrict__ xf) {
  int bt = blockIdx.x;
  int b = bt >> 12;          // / 4096
  int t = bt & (TT - 1);
  int i = t >> 6, j = t & 63;
  int c = threadIdx.x;
  const float* xb = x + (size_t)b * TT * CC;
  float w[25];
#pragma unroll
  for (int q = 0; q < 25; ++q) w[q] = eff[c * 25 + q];
  float acc = 0.f;
#pragma unroll
  for (int di = 0; di < 5; ++di) {
    int ii = i + di - 2;
    if ((unsigned)ii < (unsigned)HH) {
#pragma unroll
      for (int dj = 0; dj < 5; ++dj) {
        int jj = j + dj - 2;
        if ((unsigned)jj < (unsigned)WW)
          acc += w[di * 5 + dj] * xb[((size_t)ii * WW + jj) * CC + c];
      }
    }
  }
  xf[(size_t)bt * CC + c] = acc;
}

// ---------------------------------------------------------------------------
// fp32 WMMA helper: V_WMMA_F32_16X16X4_F32
// A frag = 16x4 f32 (2 VGPR: lanes 0-15 hold K=kb,kb+1; lanes 16-31 K=kb+2,+3)
// B frag = 4x16 (mirrored), C/D = 16x16 f32 (8 VGPR, M = r + 8*half, N = lane&15)
// ---------------------------------------------------------------------------
__device__ __forceinline__ v8f wmma4(v2f a, v2f b, v8f c) {
  return __builtin_amdgcn_wmma_f32_16x16x4_f32(false, a, false, b, (short)0, c,
                                               false, false);
}

// ---------------------------------------------------------------------------
// Kernel 3: FUSED k/v/r projection. One wave owns a 32x32 output tile for ALL
// THREE outputs: each A fragment is loaded once and hit with Wk/Wv/Wr B-frags
// -> per K-step: 8 b64 loads feed 12 WMMAs (vs 1:1 unfused). Sigmoid fused
// into the r output. N = K = CC = 256 compile-time.
// ---------------------------------------------------------------------------
__global__ __launch_bounds__(128) void gemm_kvr_fused_kernel(
    const float* __restrict__ A, const float* __restrict__ Wk,
    const float* __restrict__ Wv, const float* __restrict__ Wr,
    float* __restrict__ kout, float* __restrict__ vout,
    float* __restrict__ srout) {
  const int lane = threadIdx.x & 31;
  const int wv = threadIdx.x >> 5;  // 0..3
  const int l = lane & 15;
  const int hf = lane >> 4;
  const int m0 = blockIdx.y * 64 + (wv & 1) * 32;
  const int n0 = blockIdx.x * 64 + (wv >> 1) * 32;

  v8f accK[2][2] = {{{}, {}}, {{}, {}}};
  v8f accV[2][2] = {{{}, {}}, {{}, {}}};
  v8f accR[2][2] = {{{}, {}}, {{}, {}}};

  const float* Arow0 = A + (size_t)(m0 + l) * CC;
  const float* Arow1 = A + (size_t)(m0 + 16 + l) * CC;
  const float* Wk0 = Wk + (size_t)(n0 + l) * CC;
  const float* Wk1 = Wk + (size_t)(n0 + 16 + l) * CC;
  const float* Wv0 = Wv + (size_t)(n0 + l) * CC;
  const float* Wv1 = Wv + (size_t)(n0 + 16 + l) * CC;
  const float* Wr0 = Wr + (size_t)(n0 + l) * CC;
  const float* Wr1 = Wr + (size_t)(n0 + 16 + l) * CC;

  for (int kb = 0; kb < CC; kb += 4) {
    int kc = kb + 2 * hf;
    v2f a0 = *(const v2f*)(Arow0 + kc);
    v2f a1 = *(const v2f*)(Arow1 + kc);
    v2f bk0 = *(const v2f*)(Wk0 + kc);
    v2f bk1 = *(const v2f*)(Wk1 + kc);
    v2f bv0 = *(const v2f*)(Wv0 + kc);
    v2f bv1 = *(const v2f*)(Wv1 + kc);
    v2f br0 = *(const v2f*)(Wr0 + kc);
    v2f br1 = *(const v2f*)(Wr1 + kc);
    accK[0][0] = wmma4(a0, bk0, accK[0][0]);
    accK[0][1] = wmma4(a0, bk1, accK[0][1]);
    accK[1][0] = wmma4(a1, bk0, accK[1][0]);
    accK[1][1] = wmma4(a1, bk1, accK[1][1]);
    accV[0][0] = wmma4(a0, bv0, accV[0][0]);
    accV[0][1] = wmma4(a0, bv1, accV[0][1]);
    accV[1][0] = wmma4(a1, bv0, accV[1][0]);
    accV[1][1] = wmma4(a1, bv1, accV[1][1]);
    accR[0][0] = wmma4(a0, br0, accR[0][0]);
    accR[0][1] = wmma4(a0, br1, accR[0][1]);
    accR[1][0] = wmma4(a1, br0, accR[1][0]);
    accR[1][1] = wmma4(a1, br1, accR[1][1]);
  }

#pragma unroll
  for (int mi = 0; mi < 2; ++mi) {
#pragma unroll
    for (int ni = 0; ni < 2; ++ni) {
      int rowBase = m0 + mi * 16 + hf * 8;
      int col = n0 + ni * 16 + l;
#pragma unroll
      for (int r = 0; r < 8; ++r) {
        size_t o = (size_t)(rowBase + r) * CC + col;
        kout[o] = accK[mi][ni][r];
        vout[o] = accV[mi][ni][r];
        float rv = accR[mi][ni][r];
        srout[o] = 1.0f / (1.0f + __expf(-rv));
      }
    }
  }
}

// ---------------------------------------------------------------------------
// Kernel 4: generic fp32 WMMA GEMM  Out[M,N] = A[M,K] * W[N,K]^T
// Wave tile 32x64 (2 M-frags x 4 N-frags): 6 b64 loads feed 8 WMMAs per step.
// Block = 4 waves (2M x 2N) -> 64x128 tile. Requires M%64==0, N%128==0, K%4==0.
// ---------------------------------------------------------------------------
__global__ __launch_bounds__(128) void gemm_wmma_f32_kernel(
    const float* __restrict__ A, const float* __restrict__ W,
    float* __restrict__ Out, int M, int N, int K) {
  const int lane = threadIdx.x & 31;
  const int wv = threadIdx.x >> 5;  // 0..3
  const int l = lane & 15;
  const int hf = lane >> 4;
  const int m0 = blockIdx.y * 64 + (wv & 1) * 32;
  const int n0 = blockIdx.x * 128 + (wv >> 1) * 64;

  v8f acc[2][4] = {{{}, {}, {}, {}}, {{}, {}, {}, {}}};

  const float* Arow0 = A + (size_t)(m0 + l) * K;
  const float* Arow1 = A + (size_t)(m0 + 16 + l) * K;
  const float* Wrow[4];
#pragma unroll
  for (int ni = 0; ni < 4; ++ni) Wrow[ni] = W + (size_t)(n0 + ni * 16 + l) * K;

  for (int kb = 0; kb < K; kb += 4) {
    int kc = kb + 2 * hf;
    v2f a0 = *(const v2f*)(Arow0 + kc);
    v2f a1 = *(const v2f*)(Arow1 + kc);
    v2f b0 = *(const v2f*)(Wrow[0] + kc);
    v2f b1 = *(const v2f*)(Wrow[1] + kc);
    v2f b2 = *(const v2f*)(Wrow[2] + kc);
    v2f b3 = *(const v2f*)(Wrow[3] + kc);
    acc[0][0] = wmma4(a0, b0, acc[0][0]);
    acc[0][1] = wmma4(a0, b1, acc[0][1]);
    acc[0][2] = wmma4(a0, b2, acc[0][2]);
    acc[0][3] = wmma4(a0, b3, acc[0][3]);
    acc[1][0] = wmma4(a1, b0, acc[1][0]);
    acc[1][1] = wmma4(a1, b1, acc[1][1]);
    acc[1][2] = wmma4(a1, b2, acc[1][2]);
    acc[1][3] = wmma4(a1, b3, acc[1][3]);
  }

#pragma unroll
  for (int mi = 0; mi < 2; ++mi) {
#pragma unroll
    for (int ni = 0; ni < 4; ++ni) {
      int rowBase = m0 + mi * 16 + hf * 8;
      int col = n0 + ni * 16 + l;
#pragma unroll
      for (int r = 0; r < 8; ++r)
        Out[(size_t)(rowBase + r) * N + col] = acc[mi][ni][r];
    }
  }
}

// ---------------------------------------------------------------------------
// WKV parallel scan.  State triple (a,b,pp) represents (a*e^pp, b*e^pp).
// Recurrence:  s_t = e^{w} s_{t-1} + e^{k_t} (v_t, 1)   -- associative.
// Channel d < 256 reads time index t; d >= 256 reads the W-major reordering
// tk = (t%64)*64 + t/64 of the same k/v (this IS the concat second half).
// ---------------------------------------------------------------------------
__device__ __forceinline__ int perm_t(int t, bool second) {
  return second ? (((t & 63) << 6) | (t >> 6)) : t;
}

// Phase 1: per-chunk injection triple (zero-state replay of tail update).
__global__ __launch_bounds__(DD) void wkv_phase1_kernel(
    const float* __restrict__ k, const float* __restrict__ v,
    const float* __restrict__ sdecay, float* __restrict__ injA,
    float* __restrict__ injB, float* __restrict__ injP) {
  int d = threadIdx.x;
  int b = blockIdx.x >> 6;
  int ch = blockIdx.x & (NCHUNK - 1);
  int c = d & (CC - 1);
  bool second = d >= CC;
  float wd = -__expf(sdecay[d] * (1.0f / (float)TT));
  const float* kb = k + (size_t)b * TT * CC;
  const float* vb = v + (size_t)b * TT * CC;
  float a = 0.f, bb = 0.f, pp = -1e38f;
  int t0 = ch * CLEN;
  for (int s = 0; s < CLEN; ++s) {
    int tk = perm_t(t0 + s, second);
    float kt = kb[(size_t)tk * CC + c];
    float vt = vb[(size_t)tk * CC + c];
    float ww2 = pp + wd;
    float p2 = fmaxf(ww2, kt);
    float f1 = __expf(ww2 - p2);
    float f2 = __expf(kt - p2);
    a = f1 * a + f2 * vt;
    bb = f1 * bb + f2;
    pp = p2;
  }
  size_t idx = ((size_t)b * NCHUNK + ch) * DD + d;
  injA[idx] = a;
  injB[idx] = bb;
  injP[idx] = pp;
}

// Phase 2: serial exclusive scan over the 64 chunk summaries per (b,d).
// Decaying a state by a whole chunk is just pp += CLEN*wd.
__global__ __launch_bounds__(DD) void wkv_phase2_kernel(
    const float* __restrict__ injA, const float* __restrict__ injB,
    const float* __restrict__ injP, const float* __restrict__ sdecay,
    float* __restrict__ preA, float* __restrict__ preB,
    float* __restrict__ preP) {
  int d = threadIdx.x;
  int b = blockIdx.x;
  float wd = -__expf(sdecay[d] * (1.0f / (float)TT));
  float Ldec = (float)CLEN * wd;
  float a = 0.f, bb = 0.f, pp = -1e38f;
  for (int ch = 0; ch < NCHUNK; ++ch) {
    size_t idx = ((size_t)b * NCHUNK + ch) * DD + d;
    preA[idx] = a;   // state entering chunk ch
    preB[idx] = bb;
    preP[idx] = pp;
    float ppd = pp + Ldec;
    float ai = injA[idx], bi = injB[idx], pi = injP[idx];
    float p2 = fmaxf(ppd, pi);
    float e1 = __expf(ppd - p2);
    float e2 = __expf(pi - p2);
    a = e1 * a + e2 * ai;
    bb = e1 * bb + e2 * bi;
    pp = p2;
  }
}

// Phase 3: replay each chunk from its prefix state, emit out * sigmoid(r).
__global__ __launch_bounds__(DD) void wkv_phase3_kernel(
    const float* __restrict__ k, const float* __restrict__ v,
    const float* __restrict__ sr, const float* __restrict__ sdecay,
    const float* __restrict__ sfirst, const float* __restrict__ preA,
    const float* __restrict__ preB, const float* __restrict__ preP,
    float* __restrict__ xs) {
  int d = threadIdx.x;
  int b = blockIdx.x >> 6;
  int ch = blockIdx.x & (NCHUNK - 1);
  int c = d & (CC - 1);
  bool second = d >= CC;
  float wd = -__expf(sdecay[d] * (1.0f / (float)TT));
  float u = sfirst[d] * (1.0f / (float)TT);
  size_t idx = ((size_t)b * NCHUNK + ch) * DD + d;
  float a = preA[idx], bb = preB[idx], pp = preP[idx];
  const float* kb = k + (size_t)b * TT * CC;
  const float* vb = v + (size_t)b * TT * CC;
  const float* srb = sr + (size_t)b * TT * CC;
  float* xsb = xs + (size_t)b * TT * DD;
  int t0 = ch * CLEN;
  for (int s = 0; s < CLEN; ++s) {
    int t = t0 + s;
    int tk = perm_t(t, second);
    float kt = kb[(size_t)tk * CC + c];
    float vt = vb[(size_t)tk * CC + c];
    // output (uses u)
    float ww = u + kt;
    float p = fmaxf(pp, ww);
    float e1 = __expf(pp - p);
    float e2 = __expf(ww - p);
    float out = (e1 * a + e2 * vt) / (e1 * bb + e2);
    // state update (uses w)
    float ww2 = pp + wd;
    float p2 = fmaxf(ww2, kt);
    float f1 = __expf(ww2 - p2);
    float f2 = __expf(kt - p2);
    a = f1 * a + f2 * vt;
    bb = f1 * bb + f2;
    pp = p2;
    // srs is NOT time-permuted (concat of sr with itself along channels)
    float srv = srb[(size_t)t * CC + c];
    xsb[(size_t)t * DD + d] = out * srv;
  }
}

// ---------------------------------------------------------------------------
// Host launch
// ---------------------------------------------------------------------------
extern "C" void kernel_launch(void* const* d_in, const int* in_sizes, int n_in,
                              void* d_out, int out_size, void* d_ws,
                              size_t ws_size, hipStream_t stream) {
  const float* x      = (const float*)d_in[0];
  const float* alpha  = (const float*)d_in[1];
  const float* cw1    = (const float*)d_in[2];
  const float* cw3    = (const float*)d_in[3];
  const float* cw5    = (const float*)d_in[4];
  const float* Wk     = (const float*)d_in[5];
  const float* Wv     = (const float*)d_in[6];
  const float* Wr     = (const float*)d_in[7];
  const float* Wo     = (const float*)d_in[8];
  const float* sdecay = (const float*)d_in[9];
  const float* sfirst = (const float*)d_in[10];
  float* out = (float*)d_out;

  // Workspace layout (float offsets)
  float* ws = (float*)d_ws;
  const size_t M = (size_t)BATCH * TT;        // 32768
  const size_t szBTC = M * CC;                // 8,388,608
  const size_t szBTD = M * DD;                // 16,777,216
  const size_t szInj = (size_t)BATCH * NCHUNK * DD;  // 262,144

  float* eff   = ws;                 // 6400 (pad to 8192)
  float* xf    = eff + 8192;
  float* kbuf  = xf + szBTC;
  float* vbuf  = kbuf + szBTC;
  float* srbuf = vbuf + szBTC;
  float* xsbuf = srbuf + szBTC;
  float* injA  = xsbuf + szBTD;
  float* injB  = injA + szInj;
  float* injP  = injB + szInj;
  float* preA  = injP + szInj;
  float* preB  = preA + szInj;
  float* preP  = preB + szInj;

  // 1) fold OmniShift into one effective 5x5 depthwise kernel
  build_eff_kernel<<<1, CC, 0, stream>>>(alpha, cw1, cw3, cw5, eff);

  // 2) depthwise conv -> xf [B*T, C]
  omni_conv_kernel<<<BATCH * TT, CC, 0, stream>>>(x, eff, xf);

  // 3) fused k/v/r projections via fp32 WMMA (A loaded once for 3 GEMMs,
  //    sigmoid fused into r)
  dim3 gk(CC / 64, (unsigned)(M / 64));  // (4, 512)
  gemm_kvr_fused_kernel<<<gk, 128, 0, stream>>>(xf, Wk, Wv, Wr, kbuf, vbuf,
                                                srbuf);

  // 4) WKV bidirectional recurrence as a 3-phase chunked parallel scan
  wkv_phase1_kernel<<<BATCH * NCHUNK, DD, 0, stream>>>(kbuf, vbuf, sdecay,
                                                       injA, injB, injP);
  wkv_phase2_kernel<<<BATCH, DD, 0, stream>>>(injA, injB, injP, sdecay,
                                              preA, preB, preP);
  wkv_phase3_kernel<<<BATCH * NCHUNK, DD, 0, stream>>>(
      kbuf, vbuf, srbuf, sdecay, sfirst, preA, preB, preP, xsbuf);

  // 5) output projection: out = (xs * srs) @ Wo^T   [32768,512]x[512,256]
  dim3 go(CC / 128, (unsigned)(M / 64));  // (2, 512)
  gemm_wmma_f32_kernel<<<go, 128, 0, stream>>>(xsbuf, Wo, out, (int)M, CC, DD);
}